// YoloNMS_83451214561626
// MI455X (gfx1250) — compile-verified
//
#include <hip/hip_runtime.h>
#include <hip/hip_bf16.h>
#include <stdint.h>

// Problem constants (match reference)
#define NPRED   100800
#define NCH     117        // 4 box + 1 obj + 80 cls + 32 masks
#define NCLS    80
#define NMASK   32
#define TOPK    4096
#define NPAD    131072     // 2^17 >= NPRED, bitonic size
#define MAXDET  300
#define CONF_T  0.25f
#define IOU_T   0.45f
#define NTILE   (TOPK / 16)        // 256 column tiles
#define OUT_TOTAL (MAXDET*4 + MAXDET + MAXDET + MAXDET*NMASK)  // 11400

typedef __attribute__((ext_vector_type(2))) float v2f;
typedef __attribute__((ext_vector_type(8))) float v8f;

// ---------------------------------------------------------------- K0: zero out
__global__ void k_zero(float* __restrict__ out, int n) {
    int i = blockIdx.x * blockDim.x + threadIdx.x;
    if (i < n) out[i] = 0.0f;
}

// --------------------------- K1: scores, argmax class, boxes (wave32 per row)
// Lanes cooperatively load channels 0..95 (coalesced), reduce max(cls*obj)
// with first-max (smallest class index) tie-break via shfl_xor.
__global__ void __launch_bounds__(256)
k_scores(const float* __restrict__ pred,
         float* __restrict__ scoreN,
         float* __restrict__ clsN,
         float4* __restrict__ boxesN) {
    const int gtid = blockIdx.x * blockDim.x + threadIdx.x;
    const int row  = gtid >> 5;          // one wave per prediction
    const int lane = threadIdx.x & 31;
    if (row >= NPRED) return;            // wave-uniform

    const float* p = pred + (size_t)row * NCH;
    float v0 = p[lane];                  // ch 0..31
    float v1 = p[32 + lane];             // ch 32..63
    float v2 = p[64 + lane];             // ch 64..95 (classes end at 84)

    float obj = __shfl(v0, 4, 32);

    // per-lane partial max over its classes (ascending class order; strict >)
    float bv = -3.4e38f;
    int   bc = 1 << 20;
    if (lane >= 5) { bv = v0 * obj; bc = lane - 5; }           // cls 0..26
    {
        float t = v1 * obj;                                    // cls 27..58
        if (t > bv) { bv = t; bc = lane + 27; }
    }
    if (lane < 21) {
        float t = v2 * obj;                                    // cls 59..79
        if (t > bv) { bv = t; bc = lane + 59; }
    }
    // wave reduction: max value, smallest class index on ties
    #pragma unroll
    for (int off = 16; off >= 1; off >>= 1) {
        float ov = __shfl_xor(bv, off, 32);
        int   oc = __shfl_xor(bc, off, 32);
        if (ov > bv || (ov == bv && oc < bc)) { bv = ov; bc = oc; }
    }

    float x = __shfl(v0, 0, 32);
    float y = __shfl(v0, 1, 32);
    float w = __shfl(v0, 2, 32);
    float h = __shfl(v0, 3, 32);

    if (lane == 0) {
        scoreN[row] = (obj > CONF_T) ? bv : 0.0f;
        clsN[row]   = (float)bc;
        float4 b;
        b.x = y - h * 0.5f;   // y1
        b.y = x - w * 0.5f;   // x1
        b.z = y + h * 0.5f;   // y2
        b.w = x + w * 0.5f;   // x2
        boxesN[row] = b;
    }
}

// ---------------------------------------------------------- K2: build sort keys
// key = score_bits (scores >= 0, so float bits are order-preserving) in high 32,
// (0xFFFFFFFF - idx) in low 32 -> descending sort gives score desc, idx asc ties.
__global__ void k_keys(const float* __restrict__ scoreN,
                       unsigned long long* __restrict__ keys) {
    int i = blockIdx.x * blockDim.x + threadIdx.x;
    if (i >= NPAD) return;
    if (i < NPRED) {
        unsigned b = __float_as_uint(scoreN[i]);
        keys[i] = ((unsigned long long)b << 32) | (unsigned)(0xFFFFFFFFu - (unsigned)i);
    } else {
        keys[i] = 0ull;   // pads sort last
    }
}

// ------------------------------------------------ K3: one bitonic step (descending)
__global__ void k_bitonic(unsigned long long* __restrict__ keys,
                          unsigned j, unsigned k) {
    unsigned i = blockIdx.x * blockDim.x + threadIdx.x;
    unsigned ixj = i ^ j;
    if (ixj > i && i < NPAD) {
        unsigned long long a = keys[i];
        unsigned long long b = keys[ixj];
        bool up = ((i & k) == 0);
        if (up ? (a < b) : (a > b)) { keys[i] = b; keys[ixj] = a; }
    }
}

// ------------------------------- K4: gather top-K indices, scores, boxes, areas
__global__ void k_gather(const unsigned long long* __restrict__ keys,
                         const float4* __restrict__ boxesN,
                         unsigned* __restrict__ top_i,
                         float* __restrict__ top_s,
                         float4* __restrict__ boxes4k,
                         float* __restrict__ area4k) {
    int i = blockIdx.x * blockDim.x + threadIdx.x;
    if (i >= TOPK) return;
    unsigned long long key = keys[i];
    unsigned idx = 0xFFFFFFFFu - (unsigned)(key & 0xFFFFFFFFull);
    float s = __uint_as_float((unsigned)(key >> 32));
    if (idx >= NPRED) { idx = 0; s = 0.0f; }
    top_i[i] = idx;
    top_s[i] = s;
    float4 b = boxesN[idx];
    boxes4k[i] = b;
    area4k[i] = (b.z - b.x) * (b.w - b.y);   // (y2-y1)*(x2-x1)
}

// ------------------- K5: 16x16 suppression tiles. WMMA computes area_i + area_j
// One wave32 per tile. blockIdx.x = tj (col tile), blockIdx.y = ti (row tile).
// sup[row][tj] is a u32 holding 16 suppression bits for columns tj*16..tj*16+15.
__global__ void __launch_bounds__(32)
k_suptiles(const float4* __restrict__ boxes4k,
           const float*  __restrict__ area4k,
           unsigned* __restrict__ sup) {
    const int tj = blockIdx.x;
    const int ti = blockIdx.y;
    const int lane = threadIdx.x & 31;
    const int i0 = ti * 16;
    const int j0 = tj * 16;

    if (tj < ti) {  // whole tile has j < i -> no bits (uniform branch, EXEC intact)
        if (lane < 16) sup[(size_t)(i0 + lane) * NTILE + tj] = 0u;
        return;
    }

    const int  col = lane & 15;
    const bool lo  = lane < 16;

    // A (16x4 f32): A[m][0]=area_i[m], A[m][1]=1, A[m][2..3]=0
    // lanes 0-15 hold K=0,1 ; lanes 16-31 hold K=2,3 (zeros)
    float ai = area4k[i0 + col];
    float aj = area4k[j0 + col];
    v2f a, b;
    a.x = lo ? ai   : 0.0f;  a.y = lo ? 1.0f : 0.0f;
    // B (4x16 f32): B[0][n]=1, B[1][n]=area_j[n], B[2..3][n]=0
    b.x = lo ? 1.0f : 0.0f;  b.y = lo ? aj   : 0.0f;

    v8f c = {};
    // D[m][n] = area_i[m] + area_j[n]  (rank-2 outer product on the matrix core)
    c = __builtin_amdgcn_wmma_f32_16x16x4_f32(
        /*neg_a=*/false, a, /*neg_b=*/false, b,
        /*c_mod=*/(short)0, c, /*reuse_a=*/false, /*reuse_b=*/false);

    // Column box for this lane (D element: M = r or r+8, N = lane&15)
    float4 bj = boxes4k[j0 + col];
    const int jcol = j0 + col;

    unsigned rowmask = 0u;
    #pragma unroll
    for (int r = 0; r < 8; ++r) {
        int m = lo ? r : (r + 8);
        int irow = i0 + m;
        float4 bi = boxes4k[irow];
        float iy1 = fmaxf(bi.x, bj.x);
        float ix1 = fmaxf(bi.y, bj.y);
        float iy2 = fminf(bi.z, bj.z);
        float ix2 = fminf(bi.w, bj.w);
        float inter = fmaxf(iy2 - iy1, 0.0f) * fmaxf(ix2 - ix1, 0.0f);
        float uni   = c[r] - inter;                 // areaSum from WMMA - inter
        // iou > T  <=>  inter > T * max(uni, eps)   (divide-free comparison)
        bool  s     = (inter > IOU_T * fmaxf(uni, 1e-9f)) && (jcol > irow);
        unsigned bal = __builtin_amdgcn_ballot_w32(s);
        // ballot bit l: lane l's predicate. low 16 = row r, high 16 = row r+8.
        if (lane == r)      rowmask = bal & 0xFFFFu;
        if (lane == r + 8)  rowmask = bal >> 16;
    }
    if (lane < 16) sup[(size_t)(i0 + lane) * NTILE + tj] = rowmask;
}

// ---------------------------- K6: sequential greedy NMS scan (one wave32 block)
__global__ void __launch_bounds__(32)
k_nms(const unsigned* __restrict__ sup,
      const float* __restrict__ top_s,
      int* __restrict__ sel) {
    __shared__ unsigned rem[NTILE];
    const int lane = threadIdx.x;
    for (int t = lane; t < NTILE; t += 32) rem[t] = 0u;
    for (int s = lane; s < MAXDET; s += 32) sel[s] = -1;
    __syncthreads();

    int cnt = 0;
    for (int i = 0; i < TOPK; ++i) {
        // hide the serial row-fetch latency (global_prefetch_b8)
        if (i + 1 < TOPK)
            __builtin_prefetch(sup + (size_t)(i + 1) * NTILE + lane * 8, 0, 1);

        unsigned word = rem[i >> 4];                      // broadcast read
        bool alive = (top_s[i] > 0.0f) && !((word >> (i & 15)) & 1u);
        if (alive) {                                      // wave-uniform
            const unsigned* row = sup + (size_t)i * NTILE;
            for (int t = lane; t < NTILE; t += 32) rem[t] |= row[t];
        }
        if (lane == 0 && alive) {
            if (cnt < MAXDET) sel[cnt] = i;
            ++cnt;
        }
        __syncthreads();
    }
}

// ----------------------------------------- K7: write selected detections to out
// out layout: boxes[300*4] | cls[300] | scores[300] | masks[300*32]
__global__ void __launch_bounds__(32)
k_write(const int* __restrict__ sel,
        const unsigned* __restrict__ top_i,
        const float* __restrict__ top_s,
        const float* __restrict__ boxesN,
        const float* __restrict__ clsN,
        const float* __restrict__ pred,
        float* __restrict__ out) {
    int slot = blockIdx.x;
    int lane = threadIdx.x;
    int s = sel[slot];
    if (s < 0) return;            // out already zeroed -> ok==0 rows stay zero
    unsigned orig = top_i[s];
    if (lane < 4) out[slot * 4 + lane] = boxesN[(size_t)orig * 4 + lane];
    if (lane == 0) {
        out[MAXDET * 4 + slot]           = clsN[orig];
        out[MAXDET * 4 + MAXDET + slot]  = top_s[s];
    }
    out[MAXDET * 6 + slot * NMASK + lane] = pred[(size_t)orig * NCH + 5 + NCLS + lane];
}

// ------------------------------------------------------------------- dispatch
static inline size_t align_up(size_t v, size_t a) { return (v + a - 1) & ~(a - 1); }

extern "C" void kernel_launch(void* const* d_in, const int* in_sizes, int n_in,
                              void* d_out, int out_size, void* d_ws, size_t ws_size,
                              hipStream_t stream) {
    const float* pred = (const float*)d_in[0];
    float* out = (float*)d_out;

    char* base = (char*)d_ws;
    size_t off = 0;
    auto carve = [&](size_t bytes) -> char* {
        off = align_up(off, 256);
        char* p = base + off;
        off += bytes;
        return p;
    };
    float*              scoreN  = (float*)carve(sizeof(float) * NPRED);
    float*              clsN    = (float*)carve(sizeof(float) * NPRED);
    float4*             boxesN  = (float4*)carve(sizeof(float4) * NPRED);
    unsigned long long* keys    = (unsigned long long*)carve(sizeof(unsigned long long) * NPAD);
    unsigned*           top_i   = (unsigned*)carve(sizeof(unsigned) * TOPK);
    float*              top_s   = (float*)carve(sizeof(float) * TOPK);
    float4*             boxes4k = (float4*)carve(sizeof(float4) * TOPK);
    float*              area4k  = (float*)carve(sizeof(float) * TOPK);
    unsigned*           sup     = (unsigned*)carve(sizeof(unsigned) * TOPK * NTILE);
    int*                sel     = (int*)carve(sizeof(int) * MAXDET);
    (void)ws_size; (void)in_sizes; (void)n_in; (void)out_size;

    // K0: zero the output (suppressed/empty slots must be exactly 0)
    k_zero<<<(OUT_TOTAL + 255) / 256, 256, 0, stream>>>(out, OUT_TOTAL);

    // K1: scores / class argmax / yxyx boxes (wave per row, coalesced)
    k_scores<<<(NPRED * 32 + 255) / 256, 256, 0, stream>>>(pred, scoreN, clsN, boxesN);

    // K2: sort keys
    k_keys<<<(NPAD + 255) / 256, 256, 0, stream>>>(scoreN, keys);

    // K3: bitonic sort, descending (153 steps, keys stay L2-resident)
    for (unsigned k = 2; k <= NPAD; k <<= 1)
        for (unsigned j = k >> 1; j >= 1; j >>= 1)
            k_bitonic<<<NPAD / 256, 256, 0, stream>>>(keys, j, k);

    // K4: gather top-K
    k_gather<<<(TOPK + 255) / 256, 256, 0, stream>>>(keys, boxesN, top_i, top_s,
                                                     boxes4k, area4k);

    // K5: suppression bitmask tiles (WMMA area-sum + VALU intersection + ballot)
    dim3 supgrid(NTILE, NTILE);
    k_suptiles<<<supgrid, 32, 0, stream>>>(boxes4k, area4k, sup);

    // K6: serial greedy NMS scan, emits selection order
    k_nms<<<1, 32, 0, stream>>>(sup, top_s, sel);

    // K7: scatter selected detections
    k_write<<<MAXDET, 32, 0, stream>>>(sel, top_i, top_s, (const float*)boxesN,
                                       clsN, pred, out);
}